// GAT_Decoder_7438883357100
// MI455X (gfx1250) — compile-verified
//
#include <hip/hip_runtime.h>
#include <math.h>

#define B_ 512
#define N_ 101
#define H_ 128
#define NH_ 8
#define HD_ 16
#define NEG_ (-1e9f)
#define EPS_ (1e-5f)

typedef __attribute__((ext_vector_type(2))) float v2f;
typedef __attribute__((ext_vector_type(4))) float v4f;
typedef __attribute__((ext_vector_type(8))) float v8f;

static __device__ __forceinline__ v8f wmma_f32x4(v2f a, v2f b, v8f c) {
  // V_WMMA_F32_16X16X4_F32: D(16x16,f32) = A(16x4,f32) * B(4x16,f32) + C
  return __builtin_amdgcn_wmma_f32_16x16x4_f32(false, a, false, b, (short)0, c,
                                               false, false);
}

// -------------------------------------------------------------------------
// Kernel 1: K_m / V_m / K_p projections:  Out[b,n,:] = E[b,n,:] @ W^T
// grid = (B, 7, 3); block = 256 (8 waves, one 16-col tile each).
// fp32 WMMA 16x16x4, K=128 fully unrolled (32 wmma per wave).
// -------------------------------------------------------------------------
__global__ void proj_kernel(const float* __restrict__ E,
                            const float* __restrict__ Wk,
                            const float* __restrict__ Wv,
                            const float* __restrict__ Wp,
                            float* __restrict__ Km,
                            float* __restrict__ Vm,
                            float* __restrict__ Kp) {
  const int b = blockIdx.x;
  const int m0 = blockIdx.y * 16;
  const int which = blockIdx.z;
  const float* W = (which == 0) ? Wk : ((which == 1) ? Wv : Wp);
  float* Out = (which == 0) ? Km : ((which == 1) ? Vm : Kp);

  const int wave = threadIdx.x >> 5;
  const int lane = threadIdx.x & 31;
  const int hf = lane >> 4;
  const int l16 = lane & 15;
  const int n0 = wave * 16;

  const float* Eb = E + (size_t)b * N_ * H_;
  const int row = m0 + l16;
  const bool rowok = row < N_;
  const int rowc = rowok ? row : 0;  // clamp; results for fake rows discarded
  const float* arow = Eb + (size_t)rowc * H_ + hf * 2;
  const float* brow = W + (size_t)(n0 + l16) * H_ + hf * 2;

  v8f acc = {0.f, 0.f, 0.f, 0.f, 0.f, 0.f, 0.f, 0.f};
#pragma unroll
  for (int k = 0; k < H_; k += 4) {
    v2f a = *(const v2f*)(arow + k);
    v2f b2 = *(const v2f*)(brow + k);
    acc = wmma_f32x4(a, b2, acc);
  }

  float* ob = Out + (size_t)b * N_ * H_ + n0 + l16;
#pragma unroll
  for (int i = 0; i < 8; ++i) {
    int m = m0 + i + hf * 8;
    if (m < N_) ob[(size_t)m * H_] = acc[i];
  }
}

// -------------------------------------------------------------------------
// Kernel 2: dec_in, Q_m, compat (small). One block per batch row.
// -------------------------------------------------------------------------
__global__ void prep_kernel(const float* __restrict__ E,
                            const float* __restrict__ pool,
                            const float* __restrict__ cap_p,
                            const float* __restrict__ Wfc,   // (H, H+1)
                            const float* __restrict__ Wfc1,  // (H, H)
                            const float* __restrict__ Ww,    // (H, H)
                            const float* __restrict__ Km,    // (B, N, H)
                            float* __restrict__ compat) {    // (B, NH, N)
  __shared__ float sdec[H_];
  __shared__ float sq[H_];
  const int b = blockIdx.x;
  const int h = threadIdx.x;

  const float* e0 = E + (size_t)b * N_ * H_;
  const float* pb = pool + (size_t)b * H_;
  const float cap = cap_p[0];

  float acc = cap * Wfc[(size_t)h * (H_ + 1) + H_];
  for (int j = 0; j < H_; ++j) acc += e0[j] * Wfc[(size_t)h * (H_ + 1) + j];
  for (int j = 0; j < H_; ++j) acc += pb[j] * Wfc1[(size_t)h * H_ + j];
  sdec[h] = acc;
  __syncthreads();

  float q = 0.0f;
  for (int j = 0; j < H_; ++j) q += sdec[j] * Ww[(size_t)h * H_ + j];
  sq[h] = q;
  __syncthreads();

  const float* Kb = Km + (size_t)b * N_ * H_;
  for (int idx = h; idx < NH_ * N_; idx += H_) {
    int hh = idx / N_;
    int n = idx - hh * N_;
    float s = 0.0f;
#pragma unroll
    for (int d = 0; d < HD_; ++d)
      s += sq[hh * HD_ + d] * Kb[(size_t)n * H_ + hh * HD_ + d];
    compat[(size_t)b * NH_ * N_ + idx] = s * 0.25f;  // 1/sqrt(HD)
  }
}

// -------------------------------------------------------------------------
// Kernel 3: sequential greedy decode, ONE batch row per block (512 blocks,
// 256 threads = 8 waves). V_m[b] and K_p[b] are staged into LDS once via
// async-to-LDS (ASYNCcnt) so every per-step inner loop reads LDS only.
// Softmax / argmax / logsumexp are wave-parallel shuffle reductions.
// In-loop WMMA: glimpse = glimpse_pre(1x128, broadcast) @ W_fc_mha^T.
// -------------------------------------------------------------------------
#define VS_ 132  // padded LDS row stride (floats) for V_m / K_p tiles
#define SH_FLOATS_ (2 * N_ * VS_ + 2 * NH_ * N_ + 4 * N_ + VS_ + H_ + 8)

__global__ void decode_kernel(const float* __restrict__ Vm,
                              const float* __restrict__ Kp,
                              const float* __restrict__ Wfcm,
                              const float* __restrict__ compat,
                              const float* __restrict__ demand,
                              const float* __restrict__ cap_p,
                              const float* __restrict__ T_p,
                              const int* __restrict__ nsteps_p,
                              float* __restrict__ out) {
  extern __shared__ float smem[];
  float* s_vm     = smem;                  // [N][VS_]
  float* s_kp     = s_vm + N_ * VS_;       // [N][VS_]
  float* s_compat = s_kp + N_ * VS_;       // [NH][N]
  float* s_a      = s_compat + NH_ * N_;   // [NH][N]
  float* s_mask1  = s_a + NH_ * N_;        // [N]
  float* s_mask   = s_mask1 + N_;          // [N]
  float* s_dem    = s_mask + N_;           // [N]
  float* s_ptr    = s_dem + N_;            // [N]
  float* s_gp     = s_ptr + N_;            // [VS_] glimpse_pre
  float* s_g      = s_gp + VS_;            // [H]  glimpse
  float* s_misc   = s_g + H_;              // 0:dyn 1:logp 2:done 3:viscnt

  const int tid = threadIdx.x;
  const int b = blockIdx.x;
  const float cap = cap_p[0];
  const float invT = 1.0f / T_p[0];
  const int nsteps = nsteps_p[0];

  const int wave = tid >> 5;
  const int lane = tid & 31;
  const int hf = lane >> 4;
  const int l16 = lane & 15;
  const int n0w = wave * 16;

  // ---- stage V_m[b], K_p[b] into LDS with async-to-LDS (no VGPR transit) ----
  {
    const float* Vmb = Vm + (size_t)b * N_ * H_;
    const float* Kpb = Kp + (size_t)b * N_ * H_;
    for (int ch = tid; ch < 2 * N_ * 32; ch += 256) {
      const int t = (ch >= N_ * 32) ? 1 : 0;
      const int lc = ch - t * (N_ * 32);
      const int r = lc >> 5;          // row 0..100
      const int c4 = (lc & 31) << 2;  // float col, multiple of 4
      const float* g = (t ? Kpb : Vmb) + (size_t)r * H_ + c4;
      float* l = (t ? s_kp : s_vm) + r * VS_ + c4;
      const unsigned la = (unsigned)(uintptr_t)l;  // low 32 bits = LDS offset
      asm volatile("global_load_async_to_lds_b128 %0, %1, off"
                   :
                   : "v"(la), "v"(g)
                   : "memory");
    }
  }
  // ---- other loop-invariant state (small, plain loads) ----
  for (int i = tid; i < NH_ * N_; i += 256)
    s_compat[i] = compat[(size_t)b * NH_ * N_ + i];
  for (int i = tid; i < N_; i += 256) {
    s_dem[i] = demand[(size_t)b * N_ + i];
    s_mask1[i] = (i == 0) ? 1.0f : 0.0f;  // update_mask(dyn0, idx=0, zeros)
  }
  if (tid == 0) {
    s_misc[0] = cap;   // dyn
    s_misc[1] = 0.0f;  // logp accumulator
    s_misc[2] = 0.0f;  // done
    s_misc[3] = 0.0f;  // visited customer count
  }
  asm volatile("s_wait_asynccnt 0" ::: "memory");
  __syncthreads();

  for (int step = 0; step < nsteps; ++step) {
    // ---- done flag ----
    if (tid == 0) s_misc[2] = (s_misc[3] >= (float)(N_ - 1)) ? 1.0f : 0.0f;
    __syncthreads();
    // ---- step mask ----
    if (tid < N_) {
      float m = (s_dem[tid] > s_misc[0] - EPS_) ? 1.0f : s_mask1[tid];
      if (tid == 0 && s_misc[2] > 0.5f) m = 0.0f;
      s_mask[tid] = m;
    }
    __syncthreads();
    // ---- masked softmax, one wave per head, shuffle reductions ----
    {
      const int h = wave;
      float vl[4];
      float mx = -3.4e38f;
#pragma unroll
      for (int j = 0; j < 4; ++j) {
        int n = lane + 32 * j;
        float v = (n < N_) ? ((s_mask[n] > 0.5f) ? NEG_ : s_compat[h * N_ + n])
                           : -3.4e38f;
        vl[j] = v;
        mx = fmaxf(mx, v);
      }
#pragma unroll
      for (int m = 16; m >= 1; m >>= 1) mx = fmaxf(mx, __shfl_xor(mx, m, 32));
      float sum = 0.0f;
      float ev[4];
#pragma unroll
      for (int j = 0; j < 4; ++j) {
        int n = lane + 32 * j;
        ev[j] = (n < N_) ? expf(vl[j] - mx) : 0.0f;
        sum += ev[j];
      }
#pragma unroll
      for (int m = 16; m >= 1; m >>= 1) sum += __shfl_xor(sum, m, 32);
      const float inv = 1.0f / sum;
#pragma unroll
      for (int j = 0; j < 4; ++j) {
        int n = lane + 32 * j;
        if (n < N_) s_a[h * N_ + n] = ev[j] * inv;
      }
    }
    __syncthreads();
    // ---- glimpse_pre[c] = sum_n a[h(c)][n] * V_m[n][c]  (LDS only) ----
    if (tid < H_) {
      const int c = tid;
      const int h = c >> 4;
      const float* ar = s_a + h * N_;
      float acc = 0.0f;
      for (int n = 0; n < N_; ++n) acc += ar[n] * s_vm[n * VS_ + c];
      s_gp[c] = acc;
    }
    __syncthreads();
    // ---- glimpse = glimpse_pre @ W_fc_mha^T via WMMA (row broadcast) ----
    {
      v8f acc = {0.f, 0.f, 0.f, 0.f, 0.f, 0.f, 0.f, 0.f};
      const float* arow = s_gp + hf * 2;  // same row for all 16 A rows
      const float* brow = Wfcm + (size_t)(n0w + l16) * H_ + hf * 2;
#pragma unroll
      for (int k = 0; k < H_; k += 4) {
        v2f a = *(const v2f*)(arow + k);
        v2f b2 = *(const v2f*)(brow + k);
        acc = wmma_f32x4(a, b2, acc);
      }
      if (hf == 0) s_g[n0w + l16] = acc[0];  // D row 0, N = n0w + lane
    }
    __syncthreads();
    // ---- pointer scores (LDS only) ----
    if (tid < N_) {
      const int n = tid;
      const float* kb = s_kp + n * VS_;
      float acc = 0.0f;
#pragma unroll 4
      for (int c = 0; c < H_; c += 4) {
        acc += s_g[c] * kb[c] + s_g[c + 1] * kb[c + 1] + s_g[c + 2] * kb[c + 2] +
               s_g[c + 3] * kb[c + 3];
      }
      s_ptr[n] = acc;
    }
    __syncthreads();
    // ---- argmax + logsumexp on wave 0, then state update by lane 0 ----
    if (wave == 0) {
      const float invs = 0.08838834764831845f;  // 1/sqrt(H)
      float lgv[4];
      float best = -3.4e38f;
      int bi = 0x7fffffff;
#pragma unroll
      for (int j = 0; j < 4; ++j) {
        int n = lane + 32 * j;
        float lg = -3.4e38f;
        if (n < N_) {
          lg = (s_mask[n] > 0.5f) ? NEG_ : 10.0f * tanhf(s_ptr[n] * invs);
          lg *= invT;
        }
        lgv[j] = lg;
        if (lg > best) { best = lg; bi = n; }  // ascending j => first max in lane
      }
#pragma unroll
      for (int m = 16; m >= 1; m >>= 1) {
        float ov = __shfl_xor(best, m, 32);
        int oi = __shfl_xor(bi, m, 32);
        if (ov > best || (ov == best && oi < bi)) { best = ov; bi = oi; }
      }
      float sum = 0.0f;
#pragma unroll
      for (int j = 0; j < 4; ++j) {
        int n = lane + 32 * j;
        if (n < N_) sum += expf(lgv[j] - best);
      }
#pragma unroll
      for (int m = 16; m >= 1; m >>= 1) sum += __shfl_xor(sum, m, 32);
      if (lane == 0) {
        float logp = -logf(sum);  // lg[bi] == best
        if (s_misc[2] > 0.5f) logp = 0.0f;
        s_misc[1] += logp;
        out[(size_t)b * nsteps + step] = (float)bi;
        // update_state + update_mask
        float curd = s_dem[bi];
        s_misc[0] = (bi == 0) ? cap : (s_misc[0] - curd);
        if (bi != 0 && s_mask1[bi] < 0.5f) s_misc[3] += 1.0f;
        s_mask1[bi] = 1.0f;
        s_mask1[0] = (bi == 0) ? 1.0f : 0.0f;
      }
    }
    __syncthreads();
  }
  if (tid == 0) out[(size_t)B_ * nsteps + b] = s_misc[1];
}

// -------------------------------------------------------------------------
extern "C" void kernel_launch(void* const* d_in, const int* in_sizes, int n_in,
                              void* d_out, int out_size, void* d_ws,
                              size_t ws_size, hipStream_t stream) {
  (void)in_sizes; (void)n_in; (void)out_size; (void)ws_size;
  const float* E      = (const float*)d_in[0];   // (B,N,H)
  const float* pool   = (const float*)d_in[1];   // (B,H)
  const float* cap    = (const float*)d_in[2];   // (1,)
  const float* demand = (const float*)d_in[3];   // (B,N)
  const int*   nsteps = (const int*)d_in[4];     // scalar
  const float* T      = (const float*)d_in[5];   // scalar
  /* greedy = d_in[6] (greedy path implemented) */
  const float* Wfc    = (const float*)d_in[7];   // (H,H+1)
  const float* Wfc1   = (const float*)d_in[8];   // (H,H)
  const float* Ww     = (const float*)d_in[9];   // (H,H)
  const float* Wk     = (const float*)d_in[10];  // (H,H)
  const float* Wv     = (const float*)d_in[11];  // (H,H)
  const float* Wfcm   = (const float*)d_in[12];  // (H,H)
  const float* Wkp    = (const float*)d_in[13];  // (H,H)

  float* ws = (float*)d_ws;
  const size_t per = (size_t)B_ * N_ * H_;
  float* Km = ws;
  float* Vm = ws + per;
  float* Kp = ws + 2 * per;
  float* compat = ws + 3 * per;  // (B, NH, N)

  proj_kernel<<<dim3(B_, (N_ + 15) / 16, 3), 256, 0, stream>>>(E, Wk, Wv, Wkp,
                                                               Km, Vm, Kp);
  prep_kernel<<<B_, H_, 0, stream>>>(E, pool, cap, Wfc, Wfc1, Ww, Km, compat);
  decode_kernel<<<B_, 256, SH_FLOATS_ * sizeof(float), stream>>>(
      Vm, Kp, Wfcm, compat, demand, cap, T, nsteps, (float*)d_out);
}